// TransducerLoss_64269890617745
// MI455X (gfx1250) — compile-verified
//
#include <hip/hip_runtime.h>
#include <hip/hip_bf16.h>
#include <math.h>

// Problem constants (from reference): B=8, T=200, U=50, V=512
#define B_   8
#define T_   200
#define U_   50
#define V_   512
#define U1_  (U_ + 1)
#define NROWS (B_ * T_ * U1_)     // 81600 softmax rows
#define WAVES_PER_BLOCK 8
#define CH   100                  // DP t-chunk height (2 chunks of 100)

typedef float v2f __attribute__((ext_vector_type(2)));
typedef float v8f __attribute__((ext_vector_type(8)));

static_assert(NROWS % WAVES_PER_BLOCK == 0, "grid sizing");
static_assert(T_ % CH == 0, "chunking");

// ---------------------------------------------------------------------------
// Kernel 1: per-row log-sum-exp over V=512; emit blank_lp and emit_lp.
// One wave32 per row, 4x coalesced b128 loads (16 floats/lane).
// Cross-lane sum reduction via V_WMMA_F32_16X16X4_F32 (B = all-ones matrix ->
// D[m][n] = s_m + s_{m+16}), then 7 in-lane adds + one shfl_xor(16).
// Max reduction via 5x shfl_xor. Arguments to exp are max-subtracted (<= 0),
// so native __expf (v_exp_f32) is safe: no overflow, underflow flushes to 0.
// ---------------------------------------------------------------------------
__global__ __launch_bounds__(WAVES_PER_BLOCK * 32)
void lse_kernel(const float* __restrict__ logits,
                const int*   __restrict__ targets,
                const int*   __restrict__ blank_idx,
                float* __restrict__ blank_lp,
                float* __restrict__ emit_lp) {
  const int lane  = threadIdx.x & 31;
  const int wave  = threadIdx.x >> 5;
  const int row   = blockIdx.x * WAVES_PER_BLOCK + wave;   // < NROWS (fits in int)
  const float* base = logits + (size_t)row * V_;

  // 16 elements per lane: element index = j*128 + lane*4 + k   (fully coalesced b128)
  float4 q0 = *(const float4*)(base + 0 * 128 + lane * 4);
  float4 q1 = *(const float4*)(base + 1 * 128 + lane * 4);
  float4 q2 = *(const float4*)(base + 2 * 128 + lane * 4);
  float4 q3 = *(const float4*)(base + 3 * 128 + lane * 4);

  // lane-local max of 16
  float m = fmaxf(fmaxf(fmaxf(q0.x, q0.y), fmaxf(q0.z, q0.w)),
                  fmaxf(fmaxf(fmaxf(q1.x, q1.y), fmaxf(q1.z, q1.w)),
                        fmaxf(fmaxf(fmaxf(q2.x, q2.y), fmaxf(q2.z, q2.w)),
                              fmaxf(fmaxf(q3.x, q3.y), fmaxf(q3.z, q3.w)))));
  // wave32 max reduction
  #pragma unroll
  for (int off = 16; off >= 1; off >>= 1)
    m = fmaxf(m, __shfl_xor(m, off, 32));

  // lane-local sum of exp(x - m); args <= 0 -> native exp is safe
  float s = 0.0f;
  s += __expf(q0.x - m); s += __expf(q0.y - m); s += __expf(q0.z - m); s += __expf(q0.w - m);
  s += __expf(q1.x - m); s += __expf(q1.y - m); s += __expf(q1.z - m); s += __expf(q1.w - m);
  s += __expf(q2.x - m); s += __expf(q2.y - m); s += __expf(q2.z - m); s += __expf(q2.w - m);
  s += __expf(q3.x - m); s += __expf(q3.y - m); s += __expf(q3.z - m); s += __expf(q3.w - m);

  // --- WMMA cross-lane sum reduction (exact f32, RNE) ---
  // A (16x4 f32): VGPR0 lanes0-15 = A[m][0]=s_m, lanes16-31 = A[m][2]=s_{m+16};
  //               VGPR1 = A[m][1]/A[m][3] = 0.
  // B (4x16 f32) = all ones (layout-independent) => D[m][n] = s_m + s_{m+16}.
  v2f a;  a.x = s;     a.y = 0.0f;
  v2f bm; bm.x = 1.0f; bm.y = 1.0f;
  v8f c = {0.0f, 0.0f, 0.0f, 0.0f, 0.0f, 0.0f, 0.0f, 0.0f};
  v8f d = __builtin_amdgcn_wmma_f32_16x16x4_f32(false, a, false, bm,
                                                (short)0, c, false, false);
  // lanes 0-15 hold pair-sums of rows 0..7 in d[0..7]; lanes 16-31 rows 8..15.
  float tot = ((d[0] + d[1]) + (d[2] + d[3])) + ((d[4] + d[5]) + (d[6] + d[7]));
  tot += __shfl_xor(tot, 16, 32);   // full 32-lane sum, in every lane

  const float lse = m + logf(tot);  // keep precise log (1 per row)

  if (lane == 0) {
    const int bcol = *blank_idx;                 // == 0 for this problem
    blank_lp[row] = base[bcol] - lse;            // L0 hit (row just streamed)
    // 32-bit decomposition (row < 81600): avoids u64 magic-multiply chains
    const int u  = row % U1_;
    const int bt = row / U1_;
    const int t  = bt % T_;
    const int b  = bt / T_;
    if (u < U_) {
      const int tgt = targets[b * U_ + u];
      emit_lp[(b * T_ + t) * U_ + u] = base[tgt] - lse;   // L0 hit
    }
  }
}

// ---------------------------------------------------------------------------
// Kernel 2: alpha DP via anti-diagonal wavefront. One wave32 block per batch
// element; lane l owns columns u=2l and u=2l+1. Alpha frontier A[u] in LDS.
// blank/emit staged into LDS in t-chunks of CH rows (~41 KB).
//   alpha[0][0]=0
//   alpha[0][u]=alpha[0][u-1]+emit[0][u-1]
//   alpha[t][0]=alpha[t-1][0]+blank[t-1][0]
//   alpha[t][u]=logaddexp(alpha[t-1][u]+blank[t-1][u], alpha[t][u-1]+emit[t][u-1])
// score[b] = alpha[ti][ui] + blank[ti][ui],  ti=len_t-1, ui=len_u
// ---------------------------------------------------------------------------
__device__ __forceinline__ float logaddexpf_(float p, float q) {
  float mx = fmaxf(p, q);
  float dd = fabsf(p - q);
  return mx + log1pf(expf(-dd));
}

__global__ __launch_bounds__(32)
void dp_kernel(const float* __restrict__ blank_lp,
               const float* __restrict__ emit_lp,
               const int*   __restrict__ logit_lengths,
               const int*   __restrict__ target_lengths,
               float* __restrict__ scores) {
  __shared__ float blank_s[(CH + 1) * U1_];   // blank rows [t0-1 .. t0+CH-1] -> LDS rows 0..CH
  __shared__ float emit_s[CH * U_];           // emit  rows [t0   .. t0+CH-1]
  __shared__ float A[64];                     // alpha frontier per column u

  const int b   = blockIdx.x;
  const int tid = threadIdx.x;
  const int ti  = logit_lengths[b] - 1;
  const int ui  = target_lengths[b];
  const float* blankB = blank_lp + (size_t)b * T_ * U1_;
  const float* emitB  = emit_lp  + (size_t)b * T_ * U_;
  const int u0 = 2 * tid;
  const int u1 = 2 * tid + 1;

  for (int chunk = 0; chunk < T_ / CH; ++chunk) {
    const int t0 = chunk * CH;

    // stage blank rows [t0-1, t0+CH-1] (row t-1 -> LDS row t-t0)
    for (int i = tid; i < (CH + 1) * U1_; i += 32) {
      const int t = t0 - 1 + i / U1_;
      blank_s[i] = (t >= 0) ? blankB[t * U1_ + (i % U1_)] : 0.0f;
    }
    // stage emit rows [t0, t0+CH-1]
    for (int i = tid; i < CH * U_; i += 32) {
      emit_s[i] = emitB[t0 * U_ + i];
    }
    __syncthreads();

    for (int dg = t0; dg <= t0 + CH - 1 + U_; ++dg) {
      // read phase: snapshot old frontier values needed by this lane's 2 cells
      const float a_u0   = (u0 <= U_) ? A[u0] : 0.0f;
      const float a_u0m1 = (u0 >= 1 && u0 <= U_) ? A[u0 - 1] : 0.0f;
      const float a_u1   = (u1 <= U_) ? A[u1] : 0.0f;
      __syncthreads();

      // cell (t, u0)
      {
        const int t = dg - u0;
        if (u0 <= U_ && t >= t0 && t < t0 + CH) {
          float v;
          if (t == 0 && u0 == 0)      v = 0.0f;
          else if (t == 0)            v = a_u0m1 + emit_s[(u0 - 1)];
          else if (u0 == 0)           v = a_u0 + blank_s[(t - t0) * U1_];
          else                        v = logaddexpf_(a_u0   + blank_s[(t - t0) * U1_ + u0],
                                                      a_u0m1 + emit_s[(t - t0) * U_ + (u0 - 1)]);
          A[u0] = v;
          if (t == ti && u0 == ui)
            scores[b] = v + blank_s[(ti - t0 + 1) * U1_ + ui];
        }
      }
      // cell (t, u1): old A[u1-1] is a_u0 (same-lane snapshot)
      {
        const int t = dg - u1;
        if (u1 <= U_ && t >= t0 && t < t0 + CH) {
          float v;
          if (t == 0)                 v = a_u0 + emit_s[(u1 - 1)];       // u1 >= 1 always
          else                        v = logaddexpf_(a_u1 + blank_s[(t - t0) * U1_ + u1],
                                                      a_u0 + emit_s[(t - t0) * U_ + (u1 - 1)]);
          A[u1] = v;
          if (t == ti && u1 == ui)
            scores[b] = v + blank_s[(ti - t0 + 1) * U1_ + ui];
        }
      }
      __syncthreads();
    }
    __syncthreads();   // done with this chunk's LDS before reload
  }
}

// ---------------------------------------------------------------------------
// Kernel 3: loss = -mean(scores)
// ---------------------------------------------------------------------------
__global__ void finalize_kernel(const float* __restrict__ scores,
                                float* __restrict__ out) {
  if (threadIdx.x == 0) {
    float s = 0.0f;
    #pragma unroll
    for (int i = 0; i < B_; ++i) s += scores[i];
    out[0] = -s / (float)B_;
  }
}

// ---------------------------------------------------------------------------
extern "C" void kernel_launch(void* const* d_in, const int* in_sizes, int n_in,
                              void* d_out, int out_size, void* d_ws, size_t ws_size,
                              hipStream_t stream) {
  const float* logits         = (const float*)d_in[0];
  const int*   targets        = (const int*)d_in[1];
  const int*   logit_lengths  = (const int*)d_in[2];
  const int*   target_lengths = (const int*)d_in[3];
  const int*   blank_idx      = (const int*)d_in[4];   // scalar (==0)

  float* ws       = (float*)d_ws;
  float* blank_lp = ws;                                  // NROWS floats
  float* emit_lp  = blank_lp + NROWS;                    // B*T*U floats
  float* scores   = emit_lp + (size_t)B_ * T_ * U_;      // B floats

  lse_kernel<<<NROWS / WAVES_PER_BLOCK, WAVES_PER_BLOCK * 32, 0, stream>>>(
      logits, targets, blank_idx, blank_lp, emit_lp);
  dp_kernel<<<B_, 32, 0, stream>>>(blank_lp, emit_lp, logit_lengths,
                                   target_lengths, scores);
  finalize_kernel<<<1, 32, 0, stream>>>(scores, (float*)d_out);
}